// MultiHeadAttentionLayer_1580547970976
// MI455X (gfx1250) — compile-verified
//
#include <hip/hip_runtime.h>
#include <math.h>

#define HID 1024
#define NH  16
#define HD  64
#define NB  32
#define SKV 2048

typedef __attribute__((ext_vector_type(2))) float v2f;
typedef __attribute__((ext_vector_type(8))) float v8f;

// ---------------------------------------------------------------------------
// Wave-level 16x16 f32 tile GEMM using V_WMMA_F32_16X16X4_F32.
// A is a 16xK operand addressed as A[m*a_rs + k*a_cs]
// B is a Kx16 operand addressed as B[k*b_ks + n*b_ns]
// Fragment layout (ISA 7.12.2, 32-bit A 16x4 / mirrored B 4x16):
//   lane m=n=lane&15 ; K pair offset 2*(lane>>4)
// ---------------------------------------------------------------------------
__device__ __forceinline__ v8f wave_gemm_f32(const float* __restrict__ A, int a_rs, int a_cs,
                                             const float* __restrict__ B, int b_ks, int b_ns,
                                             int K, v8f acc)
{
    const int lane = threadIdx.x & 31;
    const int mn   = lane & 15;
    const int hi   = lane >> 4;
    const float* __restrict__ ap = A + (long)mn * a_rs;
    const float* __restrict__ bp = B + (long)mn * b_ns;
    for (int k = 0; k < K; k += 4) {
        const int k0 = k + 2 * hi;
        v2f a, b;
        a[0] = ap[(long)k0 * a_cs];
        a[1] = ap[(long)(k0 + 1) * a_cs];
        b[0] = bp[(long)k0 * b_ks];
        b[1] = bp[(long)(k0 + 1) * b_ks];
        acc = __builtin_amdgcn_wmma_f32_16x16x4_f32(false, a, false, b, (short)0, acc,
                                                    false, false);
    }
    return acc;
}

// ---------------------------------------------------------------------------
// 1) LayerNorm over q rows (32 rows of 1024, eps=1e-6)
// ---------------------------------------------------------------------------
__global__ void k_ln(const float* __restrict__ q, const float* __restrict__ w,
                     const float* __restrict__ bias, float* __restrict__ qn)
{
    const int b = blockIdx.x, t = threadIdx.x;
    const float* __restrict__ x = q + (long)b * HID;
    __shared__ float red[256];
    float v4[4];
    float s = 0.f;
#pragma unroll
    for (int i = 0; i < 4; ++i) { v4[i] = x[t + i * 256]; s += v4[i]; }
    red[t] = s; __syncthreads();
    for (int o = 128; o > 0; o >>= 1) { if (t < o) red[t] += red[t + o]; __syncthreads(); }
    const float mu = red[0] * (1.0f / HID);
    __syncthreads();
    s = 0.f;
#pragma unroll
    for (int i = 0; i < 4; ++i) { float d = v4[i] - mu; s += d * d; }
    red[t] = s; __syncthreads();
    for (int o = 128; o > 0; o >>= 1) { if (t < o) red[t] += red[t + o]; __syncthreads(); }
    const float inv = rsqrtf(red[0] * (1.0f / HID) + 1e-6f);
#pragma unroll
    for (int i = 0; i < 4; ++i) {
        const int c = t + i * 256;
        qn[(long)b * HID + c] = (v4[i] - mu) * inv * w[c] + bias[c];
    }
}

// ---------------------------------------------------------------------------
// 2) Qs = (qn @ Wq^T + bq) / sqrt(HD)   [32 x 1024], K=1024
//    128 wave tiles -> 16 blocks x 8 waves
// ---------------------------------------------------------------------------
__global__ void k_qproj(const float* __restrict__ qn, const float* __restrict__ Wq,
                        const float* __restrict__ bq, float* __restrict__ Qs)
{
    const int wid = blockIdx.x * 8 + (threadIdx.x >> 5);
    const int mi = wid >> 6, ni = wid & 63;
    v8f acc = {};
    acc = wave_gemm_f32(qn + (long)mi * 16 * HID, HID, 1,
                        Wq + (long)ni * 16 * HID, 1, HID,   // B[c][j] = Wq[j*HID + c]
                        HID, acc);
    const int lane = threadIdx.x & 31;
    const int n = lane & 15, hb = (lane >> 4) * 8;
    const int j = ni * 16 + n;
    const float bj = bq[j];
#pragma unroll
    for (int g = 0; g < 8; ++g)
        Qs[(long)(mi * 16 + hb + g) * HID + j] = (acc[g] + bj) * 0.125f;
}

// ---------------------------------------------------------------------------
// 3a) u[b,h,:] = Wk_h^T @ Qs[b, h*64:h*64+64]   per h: [32 x 1024], K=64
//     16h * 2mi * 64ni = 2048 wave tiles -> 256 blocks
// ---------------------------------------------------------------------------
__global__ void k_uproj(const float* __restrict__ Qs, const float* __restrict__ Wk,
                        float* __restrict__ u)
{
    const int wid = blockIdx.x * 8 + (threadIdx.x >> 5);
    const int h  = wid >> 7;
    const int mi = (wid >> 6) & 1;
    const int ni = wid & 63;
    v8f acc = {};
    acc = wave_gemm_f32(Qs + (long)mi * 16 * HID + h * HD, HID, 1,
                        Wk + (long)h * HD * HID + ni * 16, HID, 1, // B[d][c] = Wk[(h*64+d)*HID+c]
                        HD, acc);
    const int lane = threadIdx.x & 31;
    const int n = lane & 15, hb = (lane >> 4) * 8;
    const int c = ni * 16 + n;
#pragma unroll
    for (int g = 0; g < 8; ++g) {
        const int b = mi * 16 + hb + g;
        u[((long)b * NH + h) * HID + c] = acc[g];
    }
}

// 3b) sbias[b,h] = Qs_h . bk_h   (512 tiny dots)
__global__ void k_sbias(const float* __restrict__ Qs, const float* __restrict__ bk,
                        float* __restrict__ sb)
{
    const int idx = blockIdx.x * 256 + threadIdx.x;   // 0..511
    const int b = idx >> 4, h = idx & 15;
    const float* __restrict__ qp = Qs + (long)b * HID + h * HD;
    const float* __restrict__ bp = bk + h * HD;
    float acc = 0.f;
#pragma unroll 8
    for (int d = 0; d < HD; ++d) acc += qp[d] * bp[d];
    sb[idx] = acc;
}

// ---------------------------------------------------------------------------
// 4) scores: per b, S[s,h] = k[b,s,:] . u[b,h,:] + sbias  (M=2048,N=16,K=1024)
//    streams all of k (256 MB) exactly once. 32b*128 tiles -> 512 blocks
// ---------------------------------------------------------------------------
__global__ void k_scores(const float* __restrict__ kin, const float* __restrict__ u,
                         const float* __restrict__ sb, float* __restrict__ att)
{
    const int wid = blockIdx.x * 8 + (threadIdx.x >> 5);
    const int b  = wid >> 7;
    const int si = wid & 127;
    v8f acc = {};
    acc = wave_gemm_f32(kin + (long)b * SKV * HID + (long)si * 16 * HID, HID, 1,
                        u + (long)b * NH * HID, 1, HID,    // B[c][h] = u[(b*NH+h)*HID + c]
                        HID, acc);
    const int lane = threadIdx.x & 31;
    const int h = lane & 15, hb = (lane >> 4) * 8;
    const float bias = sb[b * NH + h];
#pragma unroll
    for (int g = 0; g < 8; ++g) {
        const int s = si * 16 + hb + g;
        att[((long)b * NH + h) * SKV + s] = acc[g] + bias;
    }
}

// ---------------------------------------------------------------------------
// 5) softmax over each att row of 2048 (in place, in d_out)
// ---------------------------------------------------------------------------
__global__ void k_softmax(float* __restrict__ att)
{
    float* __restrict__ p = att + (long)blockIdx.x * SKV;
    const int t = threadIdx.x;
    __shared__ float red[256];
    float vals[8];
    float mx = -INFINITY;
#pragma unroll
    for (int i = 0; i < 8; ++i) { vals[i] = p[t + i * 256]; mx = fmaxf(mx, vals[i]); }
    red[t] = mx; __syncthreads();
    for (int o = 128; o > 0; o >>= 1) { if (t < o) red[t] = fmaxf(red[t], red[t + o]); __syncthreads(); }
    mx = red[0]; __syncthreads();
    float s = 0.f;
#pragma unroll
    for (int i = 0; i < 8; ++i) { vals[i] = __expf(vals[i] - mx); s += vals[i]; }
    red[t] = s; __syncthreads();
    for (int o = 128; o > 0; o >>= 1) { if (t < o) red[t] += red[t + o]; __syncthreads(); }
    const float inv = 1.0f / red[0];
#pragma unroll
    for (int i = 0; i < 8; ++i) p[t + i * 256] = vals[i] * inv;
}

// ---------------------------------------------------------------------------
// 6) a[b,h,:] = att[b,h,:] @ v[b]   (M=16,N=1024,K=2048)
//    streams all of v (256 MB) exactly once. 32b*64 tiles -> 256 blocks
// ---------------------------------------------------------------------------
__global__ void k_attv(const float* __restrict__ att, const float* __restrict__ vin,
                       float* __restrict__ aout)
{
    const int wid = blockIdx.x * 8 + (threadIdx.x >> 5);
    const int b  = wid >> 6;
    const int ni = wid & 63;
    v8f acc = {};
    acc = wave_gemm_f32(att + (long)b * NH * SKV, SKV, 1,
                        vin + (long)b * SKV * HID + ni * 16, HID, 1, // B[s][c] = v[s*HID+c]
                        SKV, acc);
    const int lane = threadIdx.x & 31;
    const int n = lane & 15, hb = (lane >> 4) * 8;
#pragma unroll
    for (int g = 0; g < 8; ++g) {
        const int h = hb + g;    // M dimension == head
        aout[((long)b * NH + h) * HID + ni * 16 + n] = acc[g];
    }
}

// ---------------------------------------------------------------------------
// 7) o[b, h*64+d] = a[b,h,:] . Wv[h*64+d,:] + bv   per h: [32 x 64], K=1024
//    16h * 2mi * 4ni = 128 wave tiles -> 16 blocks
// ---------------------------------------------------------------------------
__global__ void k_oproj(const float* __restrict__ a, const float* __restrict__ Wv,
                        const float* __restrict__ bv, float* __restrict__ o)
{
    const int wid = blockIdx.x * 8 + (threadIdx.x >> 5);
    const int h  = wid >> 3;
    const int mi = (wid >> 2) & 1;
    const int ni = wid & 3;
    v8f acc = {};
    acc = wave_gemm_f32(a + (long)mi * 16 * NH * HID + h * HID, NH * HID, 1,
                        Wv + (long)(h * HD + ni * 16) * HID, 1, HID,
                        HID, acc);
    const int lane = threadIdx.x & 31;
    const int n = lane & 15, hb = (lane >> 4) * 8;
    const int d = h * HD + ni * 16 + n;
    const float bd = bv[d];
#pragma unroll
    for (int g = 0; g < 8; ++g)
        o[(long)(mi * 16 + hb + g) * HID + d] = acc[g] + bd;
}

// ---------------------------------------------------------------------------
// 8) out = o @ Wo^T + bo   [32 x 1024], K=1024 -> 128 wave tiles, 16 blocks
// ---------------------------------------------------------------------------
__global__ void k_yproj(const float* __restrict__ o, const float* __restrict__ Wo,
                        const float* __restrict__ bo, float* __restrict__ y)
{
    const int wid = blockIdx.x * 8 + (threadIdx.x >> 5);
    const int mi = wid >> 6, ni = wid & 63;
    v8f acc = {};
    acc = wave_gemm_f32(o + (long)mi * 16 * HID, HID, 1,
                        Wo + (long)ni * 16 * HID, 1, HID,
                        HID, acc);
    const int lane = threadIdx.x & 31;
    const int n = lane & 15, hb = (lane >> 4) * 8;
    const int j = ni * 16 + n;
    const float bj = bo[j];
#pragma unroll
    for (int g = 0; g < 8; ++g)
        y[(long)(mi * 16 + hb + g) * HID + j] = acc[g] + bj;
}

// ---------------------------------------------------------------------------
extern "C" void kernel_launch(void* const* d_in, const int* in_sizes, int n_in,
                              void* d_out, int out_size, void* d_ws, size_t ws_size,
                              hipStream_t stream)
{
    (void)in_sizes; (void)n_in; (void)out_size; (void)ws_size;
    const float* q    = (const float*)d_in[0];
    const float* k    = (const float*)d_in[1];
    const float* v    = (const float*)d_in[2];
    const float* ln_w = (const float*)d_in[3];
    const float* ln_b = (const float*)d_in[4];
    const float* Wq   = (const float*)d_in[5];
    const float* bq   = (const float*)d_in[6];
    const float* Wk   = (const float*)d_in[7];
    const float* bk   = (const float*)d_in[8];
    const float* Wv   = (const float*)d_in[9];
    const float* bv   = (const float*)d_in[10];
    const float* Wo   = (const float*)d_in[11];
    const float* bo   = (const float*)d_in[12];

    float* out = (float*)d_out;             // [32,1024]
    float* att = out + NB * HID;            // [32,16,2048]

    float* w    = (float*)d_ws;             // ~4.6 MB used
    float* qn   = w;                        // 32*1024
    float* Qs   = qn + NB * HID;            // 32*1024
    float* u    = Qs + NB * HID;            // 32*16*1024
    float* sb   = u + (long)NB * NH * HID;  // 512
    float* a_ws = sb + NB * NH;             // 32*16*1024
    float* o_ws = a_ws + (long)NB * NH * HID; // 32*1024

    k_ln     <<<NB,      256, 0, stream>>>(q, ln_w, ln_b, qn);
    k_qproj  <<<16,      256, 0, stream>>>(qn, Wq, bq, Qs);
    k_uproj  <<<256,     256, 0, stream>>>(Qs, Wk, u);
    k_sbias  <<<2,       256, 0, stream>>>(Qs, bk, sb);
    k_scores <<<512,     256, 0, stream>>>(k, u, sb, att);
    k_softmax<<<NB * NH, 256, 0, stream>>>(att);
    k_attv   <<<256,     256, 0, stream>>>(att, v, a_ws);
    k_oproj  <<<16,      256, 0, stream>>>(a_ws, Wv, bv, o_ws);
    k_yproj  <<<16,      256, 0, stream>>>(o_ws, Wo, bo, out);
}